// ResNetDetectionModel_89412629168322
// MI455X (gfx1250) — compile-verified
//
#include <hip/hip_runtime.h>

// ---------------------------------------------------------------------------
// YOLO-style NMS post-processing for MI455X (gfx1250, wave32).
//
// One workgroup per image (grid=128, block=1024 = 32 waves).
//   Phase A: per-box masked score -> sort keys in LDS (pad 1344 -> 2048);
//            zero the per-image suppression bit-matrix in d_ws.
//   Phase B: bitonic sort (score desc, index asc == stable argsort of -score).
//   Phase C: gather sorted offset-boxes / areas / valid flags into LDS.
//   Phase E: 16x16 IoU tiles; (area_i + area_j) via V_WMMA_F32_16X16X4_F32;
//            predicate is divide-free (inter > thr * max(union,1e-9));
//            bits packed with wave32 ballot, written as plain u16 stores
//            (unique writer per halfword -> no atomics needed).
//   Phase F: wave-0 sequential suppression scan; suppressed mask lives in
//            registers (2 words/lane), keep-bit broadcast via __shfl;
//            row loads double-buffered to hide global latency.
//   Phase G: recompute non-offset boxes, write det / class_pred / keep.
// ---------------------------------------------------------------------------

typedef __attribute__((ext_vector_type(2))) float v2f;
typedef __attribute__((ext_vector_type(8))) float v8f;

#define NBOX   1344
#define NPAD   2048
#define NCH    7
#define NTILE  84            // 1344 / 16
#define WPR    42            // u32 words per bit-matrix row (1344 / 32)
#define NPAIR  (NTILE * (NTILE + 1) / 2)   // 3570 upper-tri tiles
#define IOU_THR   0.65f
#define CONF_THRE 1.0e-4f
#define CLS_OFF   225.0f     // (MAX_DIM + 1)
#define NIMG   128

__global__ __launch_bounds__(1024)
void nms_gfx1250_kernel(const float* __restrict__ pred,
                        float* __restrict__ det,
                        int* __restrict__ clsOut,
                        unsigned char* __restrict__ keepOut,
                        unsigned* __restrict__ wsMat)
{
    __shared__ float    kscore[NPAD];
    __shared__ unsigned kidx[NPAD];
    __shared__ float    x1s[NBOX], y1s[NBOX], x2s[NBOX], y2s[NBOX], areas[NBOX];
    __shared__ unsigned valids[NBOX];
    __shared__ unsigned keepOrig[NBOX];

    const int b    = blockIdx.x;
    const int tid  = threadIdx.x;
    const int nthr = blockDim.x;
    const float* p = pred + (size_t)b * NBOX * NCH;
    unsigned* mat        = wsMat + (size_t)b * NBOX * WPR;
    unsigned short* mat16 = (unsigned short*)mat;   // halfword view, stride NTILE/row

    // ---------------- Phase A: sort keys + zero init --------------------
    for (int i = tid; i < NPAD; i += nthr) {
        float ms = -3.0e38f;          // padding sorts to the very end
        if (i < NBOX) {
            const float* r = p + i * NCH;
            __builtin_prefetch(r + NCH, 0, 1);   // global_prefetch_b8
            float obj = r[4], c0 = r[5], c1 = r[6];
            float conf  = fmaxf(c0, c1);
            float score = obj * conf;
            bool  valid = (score * conf) >= CONF_THRE;
            ms = valid ? score : -1.0e9f;        // reference's masked score
        }
        kscore[i] = ms;
        kidx[i]   = (unsigned)i;
    }
    for (int i = tid; i < NBOX * WPR; i += nthr) mat[i] = 0u;
    for (int i = tid; i < NBOX; i += nthr)       keepOrig[i] = 0u;
    __syncthreads();

    // ---------------- Phase B: bitonic sort (2048 elems, 1024 threads) ---
    // Ascending in key (-score, idx)  ==  score desc, idx asc (stable argsort).
    for (int k = 2; k <= NPAD; k <<= 1) {
        for (int j = k >> 1; j > 0; j >>= 1) {
            int i   = ((tid & ~(j - 1)) << 1) | (tid & (j - 1));
            int ixj = i | j;
            float    sa = kscore[i],  sb = kscore[ixj];
            unsigned ia = kidx[i],    ib = kidx[ixj];
            bool up   = ((i & k) == 0);
            bool agtb = (sa < sb) || (sa == sb && ia > ib);  // a "greater"
            if (agtb == up) {
                kscore[i] = sb; kscore[ixj] = sa;
                kidx[i]   = ib; kidx[ixj]   = ia;
            }
            __syncthreads();
        }
    }

    // ---------------- Phase C: gather sorted box data --------------------
    for (int ps = tid; ps < NBOX; ps += nthr) {
        unsigned i = kidx[ps];
        const float* r = p + i * NCH;
        float cx = r[0], cy = r[1], w = r[2], h = r[3];
        float obj = r[4], c0 = r[5], c1 = r[6];
        float conf = fmaxf(c0, c1);
        float off  = (c1 > c0) ? CLS_OFF : 0.0f;   // class offset for NMS
        float x1 = cx - 0.5f * w + off;
        float y1 = cy - 0.5f * h + off;
        float x2 = x1 + w;
        float y2 = y1 + h;
        x1s[ps] = x1; y1s[ps] = y1; x2s[ps] = x2; y2s[ps] = y2;
        areas[ps] = (x2 - x1) * (y2 - y1);
        float score = obj * conf;
        valids[ps] = ((score * conf) >= CONF_THRE) ? 1u : 0u;
    }
    __syncthreads();

    // ---------------- Phase E: IoU tiles, WMMA area-sum ------------------
    const int wave = tid >> 5;
    const int lane = tid & 31;
    const int nw   = nthr >> 5;

    for (int u = wave; u < NPAIR; u += nw) {
        // map linear u -> upper-triangular (ti <= tj)
        int ti = 0, rem = u;
        while (rem >= NTILE - ti) { rem -= NTILE - ti; ti++; }
        int tj = ti + rem;
        int row16 = ti << 4, col16 = tj << 4;

        // A(16x4): col0 = area_i, col1 = 1 ; B(4x16): row0 = 1, row1 = area_j
        // => D[i][j] = area_i + area_j (exact fp32, one matrix op per tile)
        v2f A = {0.0f, 0.0f}, B = {0.0f, 0.0f};
        if (lane < 16) {
            A.x = areas[row16 + lane]; A.y = 1.0f;        // K=0, K=1
            B.x = 1.0f;               B.y = areas[col16 + lane];
        }
        v8f C = {0.f, 0.f, 0.f, 0.f, 0.f, 0.f, 0.f, 0.f};
        v8f D = __builtin_amdgcn_wmma_f32_16x16x4_f32(
                    false, A, false, B, (short)0, C, false, false);

        int N = col16 + (lane & 15);               // this lane's column
        float cX1 = x1s[N], cY1 = y1s[N], cX2 = x2s[N], cY2 = y2s[N];
        int rbase = row16 + ((lane >> 4) << 3);    // M = rbase + r

        #pragma unroll
        for (int r = 0; r < 8; ++r) {
            int M = rbase + r;
            float iw = fminf(x2s[M], cX2) - fmaxf(x1s[M], cX1);
            float ih = fminf(y2s[M], cY2) - fmaxf(y1s[M], cY1);
            iw = fmaxf(iw, 0.0f); ih = fmaxf(ih, 0.0f);
            float inter = iw * ih;
            float denom = fmaxf(D[r] - inter, 1.0e-9f);  // max(union, 1e-9)
            // divide-free: inter/denom > thr  <=>  inter > thr*denom (denom>0)
            bool bit = (inter > IOU_THR * denom) && (N > M);
            unsigned bal = __builtin_amdgcn_ballot_w32(bit);
            if (bal) {   // uniform skip: most tiles produce no suppression bits
                // low 16 bits -> row (row16+r), high 16 -> row (row16+8+r).
                // Each halfword (row, tj) has exactly one writer: plain store.
                unsigned half = (lane == 0) ? (bal & 0xFFFFu) : (bal >> 16);
                if ((lane == 0 || lane == 16) && half) {
                    int rowM = row16 + r + ((lane >> 4) << 3);
                    mat16[rowM * NTILE + tj] = (unsigned short)half;
                }
            }
        }
    }
    __threadfence();        // make bit-matrix (L2) visible to wave 0
    __syncthreads();

    // ---------------- Phase F: sequential suppression scan (wave 0) ------
    if (wave == 0) {
        unsigned s0 = 0u, s1 = 0u;   // lane holds suppressed words lane, lane+32
        // double-buffered row fetch: issue row i+1 before consuming row i
        unsigned nr0 = mat[lane];
        unsigned nr1 = (lane < WPR - 32) ? mat[32 + lane] : 0u;
        for (int i = 0; i < NBOX; ++i) {
            unsigned r0 = nr0, r1 = nr1;
            if (i + 1 < NBOX) {
                nr0 = mat[(i + 1) * WPR + lane];
                nr1 = (lane < WPR - 32) ? mat[(i + 1) * WPR + 32 + lane] : 0u;
            }
            int w = i >> 5, bpos = i & 31;
            unsigned cand = (w < 32) ? s0 : s1;
            unsigned word = __shfl(cand, w & 31, 32);
            bool keep = (valids[i] != 0u) && !((word >> bpos) & 1u);
            if (keep) {                 // wave-uniform branch
                if (lane == 0) keepOrig[kidx[i]] = 1u;
                s0 |= r0;
                s1 |= r1;
            }
        }
    }
    __syncthreads();

    // ---------------- Phase G: outputs (original index order) ------------
    for (int i = tid; i < NBOX; i += nthr) {
        const float* r = p + i * NCH;
        float cx = r[0], cy = r[1], w = r[2], h = r[3];
        float obj = r[4], c0 = r[5], c1 = r[6];
        float conf = fmaxf(c0, c1);
        int   cp   = (c1 > c0) ? 1 : 0;
        float score = obj * conf;
        float x1 = cx - 0.5f * w, y1 = cy - 0.5f * h;  // NO class offset here
        float x2 = x1 + w,        y2 = y1 + h;
        unsigned kf = keepOrig[i];
        float kfl = kf ? 1.0f : 0.0f;
        size_t o = (size_t)b * NBOX + i;
        float* drow = det + o * 5;
        drow[0] = x1 * kfl; drow[1] = y1 * kfl;
        drow[2] = x2 * kfl; drow[3] = y2 * kfl;
        drow[4] = score * kfl;
        clsOut[o]  = cp;
        keepOut[o] = (unsigned char)kf;
    }
}

extern "C" void kernel_launch(void* const* d_in, const int* in_sizes, int n_in,
                              void* d_out, int out_size, void* d_ws, size_t ws_size,
                              hipStream_t stream) {
    const float* pred = (const float*)d_in[0];          // (128, 1344, 7) f32
    float* det = (float*)d_out;                          // (128, 1344, 5) f32
    int* clsOut = (int*)(det + (size_t)NIMG * NBOX * 5); // (128, 1344) i32
    unsigned char* keepOut = (unsigned char*)(clsOut + (size_t)NIMG * NBOX);
    unsigned* wsMat = (unsigned*)d_ws;                   // 128 * 1344 * 42 u32
    nms_gfx1250_kernel<<<NIMG, 1024, 0, stream>>>(pred, det, clsOut, keepOut, wsMat);
}